// SelectiveSSM_34806414967467
// MI455X (gfx1250) — compile-verified
//
#include <hip/hip_runtime.h>
#include <math.h>

// Problem constants (match reference)
#define D_INNER 2048
#define D_STATE 16
#define D_CONV  4
#define DT_RANK 128
#define T_LEN   2048
#define PROJ_W  (DT_RANK + 2 * D_STATE)   // 160
#define NCHUNK  16
#define CHUNK   (T_LEN / NCHUNK)          // 128

typedef __attribute__((ext_vector_type(2))) float v2f;
typedef __attribute__((ext_vector_type(8))) float v8f;

// ---------------------------------------------------------------------------
// Kernel 1: depthwise causal conv1d (k=4) + bias + SiLU  -> x_conv[t][d]
// ---------------------------------------------------------------------------
__global__ void conv_silu_kernel(const float* __restrict__ x,
                                 const float* __restrict__ w,   // (D,1,4)
                                 const float* __restrict__ b,   // (D,)
                                 float* __restrict__ xc) {
  int idx = blockIdx.x * blockDim.x + threadIdx.x;
  if (idx >= T_LEN * D_INNER) return;
  int t = idx / D_INNER;
  int d = idx - t * D_INNER;
  float acc = b[d];
#pragma unroll
  for (int j = 0; j < D_CONV; ++j) {
    int ts = t - (D_CONV - 1) + j;
    float xv = (ts >= 0) ? x[ts * D_INNER + d] : 0.0f;
    acc = fmaf(w[d * D_CONV + j], xv, acc);
  }
  // SiLU
  xc[idx] = acc / (1.0f + expf(-acc));
}

// ---------------------------------------------------------------------------
// Kernel 2: negA[d][n] = -exp(A_log[d][n])
// ---------------------------------------------------------------------------
__global__ void nega_kernel(const float* __restrict__ A_log,
                            float* __restrict__ negA) {
  int i = blockIdx.x * blockDim.x + threadIdx.x;
  if (i < D_INNER * D_STATE) negA[i] = -expf(A_log[i]);
}

// ---------------------------------------------------------------------------
// Kernel 3: proj[t][p] = sum_k xc[t][k] * xproj_w[p][k]
//   fp32 WMMA 16x16x4; register-blocked 16(M) x 32(N) per wave so the A
//   fragment is loaded once and reused by 2 WMMAs.
//   A lane layout: m = lane&15, k = 2*(lane>>4)+j  (contiguous float2)
//   B lane layout: n = lane&15, k = 2*(lane>>4)+j  (row of xproj_w)
//   C/D layout:    n = lane&15, m = vgpr + 8*(lane>>4)
// ---------------------------------------------------------------------------
__global__ void proj_gemm_wmma(const float* __restrict__ Xc,
                               const float* __restrict__ Wp,   // (160, 2048)
                               float* __restrict__ proj) {
  const int lane = threadIdx.x;
  const int r  = lane & 15;
  const int kb = (lane >> 4) << 1;
  const int t0 = blockIdx.x << 4;
  const int p0 = blockIdx.y << 5;            // 32 columns per wave
  const float* arow  = Xc + (t0 + r) * D_INNER + kb;
  const float* brow0 = Wp + (p0 + r) * D_INNER + kb;
  const float* brow1 = Wp + (p0 + 16 + r) * D_INNER + kb;
  v8f acc0 = {};
  v8f acc1 = {};
#pragma unroll 8
  for (int k = 0; k < D_INNER; k += 4) {
    v2f a  = *(const v2f*)(arow + k);
    v2f b0 = *(const v2f*)(brow0 + k);
    v2f b1 = *(const v2f*)(brow1 + k);
    acc0 = __builtin_amdgcn_wmma_f32_16x16x4_f32(false, a, false, b0,
                                                 (short)0, acc0, false, false);
    acc1 = __builtin_amdgcn_wmma_f32_16x16x4_f32(false, a, false, b1,
                                                 (short)0, acc1, false, false);
  }
  const int m0 = (lane >> 4) << 3;
#pragma unroll
  for (int j = 0; j < 8; ++j) {
    proj[(t0 + m0 + j) * PROJ_W + p0 + r]      = acc0[j];
    proj[(t0 + m0 + j) * PROJ_W + p0 + 16 + r] = acc1[j];
  }
}

// ---------------------------------------------------------------------------
// Kernel 4: delta[t][d] = softplus( sum_r proj[t][r]*dtproj_w[d][r] + bias[d] )
//   fp32 WMMA, register-blocked 16(M) x 64(N) per wave (4 accumulators share
//   one A fragment); K=128 fully unrolled -> 128 WMMAs per wave.
//   Fused bias + softplus epilogue.
// ---------------------------------------------------------------------------
__global__ void delta_gemm_wmma(const float* __restrict__ proj,
                                const float* __restrict__ Wd,    // (2048, 128)
                                const float* __restrict__ bias,  // (2048,)
                                float* __restrict__ delta) {
  const int lane = threadIdx.x;
  const int r  = lane & 15;
  const int kb = (lane >> 4) << 1;
  const int t0 = blockIdx.x << 4;
  const int d0 = blockIdx.y << 6;            // 64 columns per wave
  const float* arow = proj + (t0 + r) * PROJ_W + kb;   // dt_in = proj[:, :128]
  const float* brow[4];
#pragma unroll
  for (int i = 0; i < 4; ++i)
    brow[i] = Wd + (d0 + 16 * i + r) * DT_RANK + kb;

  v8f acc[4];
#pragma unroll
  for (int i = 0; i < 4; ++i) acc[i] = (v8f){};

#pragma unroll
  for (int k = 0; k < DT_RANK; k += 4) {
    v2f a = *(const v2f*)(arow + k);
#pragma unroll
    for (int i = 0; i < 4; ++i) {
      v2f bm = *(const v2f*)(brow[i] + k);
      acc[i] = __builtin_amdgcn_wmma_f32_16x16x4_f32(false, a, false, bm,
                                                     (short)0, acc[i],
                                                     false, false);
    }
  }
  const int m0 = (lane >> 4) << 3;
#pragma unroll
  for (int i = 0; i < 4; ++i) {
    const float bv = bias[d0 + 16 * i + r];
#pragma unroll
    for (int j = 0; j < 8; ++j) {
      float v  = acc[i][j] + bv;
      float sp = (v > 20.0f) ? v : log1pf(expf(v));   // softplus
      delta[(t0 + m0 + j) * D_INNER + d0 + 16 * i + r] = sp;
    }
  }
}

// ---------------------------------------------------------------------------
// Kernel 5 (phase A): per-chunk scan summaries.
//   For chunk c, channel d: ap[n] = prod_t exp(delta*negA[n]),
//   h[n] = run of h = a*h + delta*x_conv*B_in[n] starting from 0.
// ---------------------------------------------------------------------------
__global__ void scan_phaseA(const float* __restrict__ delta,
                            const float* __restrict__ xc,
                            const float* __restrict__ proj,
                            const float* __restrict__ negA,
                            float* __restrict__ aprod,
                            float* __restrict__ bsum) {
  __shared__ float Bs[CHUNK * D_STATE];
  const int tid   = threadIdx.x;
  const int d     = blockIdx.x * blockDim.x + tid;
  const int c     = blockIdx.y;
  const int tbase = c * CHUNK;
  // Stage B_in rows of this chunk into LDS (broadcast reads later).
  for (int i = tid; i < CHUNK * D_STATE; i += blockDim.x) {
    int tt = i >> 4, n = i & 15;
    Bs[i] = proj[(tbase + tt) * PROJ_W + DT_RANK + n];
  }
  __syncthreads();

  float An[D_STATE], h[D_STATE], ap[D_STATE];
#pragma unroll
  for (int n = 0; n < D_STATE; ++n) {
    An[n] = negA[d * D_STATE + n];
    h[n]  = 0.0f;
    ap[n] = 1.0f;
  }
  for (int tt = 0; tt < CHUNK; ++tt) {
    const int t    = tbase + tt;
    const float dl = delta[t * D_INNER + d];
    const float bx = dl * xc[t * D_INNER + d];
#pragma unroll
    for (int n = 0; n < D_STATE; ++n) {
      float a = expf(dl * An[n]);
      ap[n] *= a;
      h[n]   = fmaf(a, h[n], bx * Bs[tt * D_STATE + n]);
    }
  }
#pragma unroll
  for (int n = 0; n < D_STATE; ++n) {
    aprod[(c * D_INNER + d) * D_STATE + n] = ap[n];
    bsum [(c * D_INNER + d) * D_STATE + n] = h[n];
  }
}

// ---------------------------------------------------------------------------
// Kernel 6 (phase B): 16-step scan of chunk summaries -> chunk entry carries.
// ---------------------------------------------------------------------------
__global__ void scan_phaseB(const float* __restrict__ aprod,
                            const float* __restrict__ bsum,
                            float* __restrict__ carry) {
  const int dn = blockIdx.x * blockDim.x + threadIdx.x;  // (d*16 + n)
  if (dn >= D_INNER * D_STATE) return;
  float h = 0.0f;
#pragma unroll
  for (int c = 0; c < NCHUNK; ++c) {
    carry[c * (D_INNER * D_STATE) + dn] = h;
    h = fmaf(aprod[c * (D_INNER * D_STATE) + dn], h,
             bsum[c * (D_INNER * D_STATE) + dn]);
  }
}

// ---------------------------------------------------------------------------
// Kernel 7 (phase C): replay each chunk with its carry; fused y epilogue:
//   y[t][d] = sum_n h[t][d][n]*C_in[t][n] + x[t][d]*D_param[d]
// ---------------------------------------------------------------------------
__global__ void scan_phaseC(const float* __restrict__ delta,
                            const float* __restrict__ xc,
                            const float* __restrict__ proj,
                            const float* __restrict__ negA,
                            const float* __restrict__ carry,
                            const float* __restrict__ x,
                            const float* __restrict__ Dp,
                            float* __restrict__ y) {
  __shared__ float Bs[CHUNK * D_STATE];
  __shared__ float Cs[CHUNK * D_STATE];
  const int tid   = threadIdx.x;
  const int d     = blockIdx.x * blockDim.x + tid;
  const int c     = blockIdx.y;
  const int tbase = c * CHUNK;
  for (int i = tid; i < CHUNK * D_STATE; i += blockDim.x) {
    int tt = i >> 4, n = i & 15;
    Bs[i] = proj[(tbase + tt) * PROJ_W + DT_RANK + n];
    Cs[i] = proj[(tbase + tt) * PROJ_W + DT_RANK + D_STATE + n];
  }
  __syncthreads();

  float An[D_STATE], h[D_STATE];
#pragma unroll
  for (int n = 0; n < D_STATE; ++n) {
    An[n] = negA[d * D_STATE + n];
    h[n]  = carry[(c * D_INNER + d) * D_STATE + n];
  }
  const float dpv = Dp[d];
  for (int tt = 0; tt < CHUNK; ++tt) {
    const int t    = tbase + tt;
    const float dl = delta[t * D_INNER + d];
    const float bx = dl * xc[t * D_INNER + d];
    float yacc = x[t * D_INNER + d] * dpv;
#pragma unroll
    for (int n = 0; n < D_STATE; ++n) {
      float a = expf(dl * An[n]);
      h[n] = fmaf(a, h[n], bx * Bs[tt * D_STATE + n]);
      yacc = fmaf(h[n], Cs[tt * D_STATE + n], yacc);
    }
    y[t * D_INNER + d] = yacc;
  }
}

// ---------------------------------------------------------------------------
// Launch
// ---------------------------------------------------------------------------
extern "C" void kernel_launch(void* const* d_in, const int* in_sizes, int n_in,
                              void* d_out, int out_size, void* d_ws, size_t ws_size,
                              hipStream_t stream) {
  const float* x        = (const float*)d_in[0];  // (1, 2048, 2048)
  const float* conv_w   = (const float*)d_in[1];  // (2048, 1, 4)
  const float* conv_b   = (const float*)d_in[2];  // (2048,)
  const float* xproj_w  = (const float*)d_in[3];  // (160, 2048)
  const float* dtproj_w = (const float*)d_in[4];  // (2048, 128)
  const float* dtproj_b = (const float*)d_in[5];  // (2048,)
  const float* A_log    = (const float*)d_in[6];  // (2048, 16)
  const float* D_param  = (const float*)d_in[7];  // (2048,)
  float* y = (float*)d_out;                       // (1, 2048, 2048) fp32

  // Workspace layout (floats); total ~41.3 MB
  float* ws    = (float*)d_ws;
  float* xc    = ws;                                   // T*D
  float* proj  = xc    + (size_t)T_LEN * D_INNER;      // T*160
  float* delta = proj  + (size_t)T_LEN * PROJ_W;       // T*D
  float* negA  = delta + (size_t)T_LEN * D_INNER;      // D*N
  float* aprod = negA  + (size_t)D_INNER * D_STATE;    // NC*D*N
  float* bsum  = aprod + (size_t)NCHUNK * D_INNER * D_STATE;
  float* carry = bsum  + (size_t)NCHUNK * D_INNER * D_STATE;

  // 1) conv + SiLU
  conv_silu_kernel<<<(T_LEN * D_INNER + 255) / 256, 256, 0, stream>>>(
      x, conv_w, conv_b, xc);

  // 2) -exp(A_log)
  nega_kernel<<<(D_INNER * D_STATE + 255) / 256, 256, 0, stream>>>(A_log, negA);

  // 3) proj GEMM (WMMA f32): grid = (T/16, 160/32), 1 wave -> 16x32 tile
  proj_gemm_wmma<<<dim3(T_LEN / 16, PROJ_W / 32), 32, 0, stream>>>(
      xc, xproj_w, proj);

  // 4) delta GEMM (WMMA f32) + bias + softplus: grid = (T/16, D/64),
  //    1 wave -> 16x64 tile
  delta_gemm_wmma<<<dim3(T_LEN / 16, D_INNER / 64), 32, 0, stream>>>(
      proj, dtproj_w, dtproj_b, delta);

  // 5) chunked scan, phase A: (D/128) x NCHUNK blocks of 128
  scan_phaseA<<<dim3(D_INNER / 128, NCHUNK), 128, 0, stream>>>(
      delta, xc, proj, negA, aprod, bsum);

  // 6) phase B: scan summaries across chunks
  scan_phaseB<<<(D_INNER * D_STATE + 255) / 256, 256, 0, stream>>>(
      aprod, bsum, carry);

  // 7) phase C: replay with carries + fused y epilogue
  scan_phaseC<<<dim3(D_INNER / 128, NCHUNK), 128, 0, stream>>>(
      delta, xc, proj, negA, carry, x, D_param, y);
}